// GNN_SpMoE_node_47751446397458
// MI455X (gfx1250) — compile-verified
//
#include <hip/hip_runtime.h>
#include <math.h>

// ---------------------------------------------------------------------------
// GNN sparse-MoE node pipeline for MI455X (gfx1250, wave32, WMMA).
// Bottleneck is the 1.6M-edge scatter (L2-resident atomics); expert GEMMs use
// full-fp32 V_WMMA_F32_16X16X4_F32 (precision-faithful, GEMM is not the
// bottleneck). BN stats computed in a first WMMA pass (sum/sumsq atomics),
// z recomputed in a second WMMA pass fused with BN + gating + residual-relu,
// avoiding a 205MB z buffer.
// ---------------------------------------------------------------------------

#define NNODES 50000
#define DIM    256
#define NEXP   4
#define NLAYER 3
#define NEDGE  1600000
#define LDA    260        // LDS row stride (floats): bank-conflict-free, 16B aligned
#define BN_EPS 1e-5f

typedef __attribute__((ext_vector_type(2))) float v2f;
typedef __attribute__((ext_vector_type(8))) float v8f;

// ----------------------------- utility kernels -----------------------------

__global__ void zero_f(float* __restrict__ p, int n) {
  int i = blockIdx.x * blockDim.x + threadIdx.x;
  if (i < n) p[i] = 0.f;
}

__global__ void deg_kernel(const int* __restrict__ src, const int* __restrict__ dst,
                           float* __restrict__ outdeg, float* __restrict__ indeg) {
  int i = blockIdx.x * blockDim.x + threadIdx.x;
  if (i < NEDGE) {
    atomicAdd(&outdeg[src[i]], 1.f);
    atomicAdd(&indeg[dst[i]], 1.f);
  }
}

__global__ void norm_kernel(const float* __restrict__ outdeg, const float* __restrict__ indeg,
                            float* __restrict__ nsrc, float* __restrict__ ndst) {
  int i = blockIdx.x * blockDim.x + threadIdx.x;
  if (i < NNODES) {
    nsrc[i] = rsqrtf(fmaxf(outdeg[i], 1.f));
    ndst[i] = rsqrtf(fmaxf(indeg[i], 1.f));
  }
}

__global__ void embed_kernel(const int* __restrict__ x, const float* __restrict__ emb,
                             float* __restrict__ h) {
  int i = blockIdx.x * blockDim.x + threadIdx.x;
  if (i < NNODES * 64) {
    int node = i >> 6;
    int c = (i & 63) << 2;
    float4 v = *(const float4*)(emb + (size_t)x[node] * DIM + c);
    *(float4*)(h + (size_t)node * DIM + c) = v;
  }
}

// ------------------------------- gating ------------------------------------
// One wave per node: lane-parallel dot products h[n]·w_gate[:,e], butterfly
// reduce, top-2-of-4 + renormalized softmax, dense 4-wide gate row (zeros
// elsewhere so ungated experts cancel exactly as in the reference).

__global__ void gate_kernel(const float* __restrict__ h, const float* __restrict__ wg,
                            float* __restrict__ gates) {
  int gid = blockIdx.x * blockDim.x + threadIdx.x;
  int node = gid >> 5;
  int lane = gid & 31;
  if (node >= NNODES) return;
  float p0 = 0.f, p1 = 0.f, p2 = 0.f, p3 = 0.f;
  const float* hp = h + (size_t)node * DIM;
  for (int d = lane; d < DIM; d += 32) {
    float hv = hp[d];
    const float* w = wg + d * NEXP;
    p0 += hv * w[0]; p1 += hv * w[1]; p2 += hv * w[2]; p3 += hv * w[3];
  }
  for (int off = 16; off > 0; off >>= 1) {
    p0 += __shfl_xor(p0, off, 32);
    p1 += __shfl_xor(p1, off, 32);
    p2 += __shfl_xor(p2, off, 32);
    p3 += __shfl_xor(p3, off, 32);
  }
  if (lane == 0) {
    float v[4] = {p0, p1, p2, p3};
    int i1 = 0;
    for (int j = 1; j < 4; j++) if (v[j] > v[i1]) i1 = j;
    int i2 = -1;
    for (int j = 0; j < 4; j++) { if (j == i1) continue; if (i2 < 0 || v[j] > v[i2]) i2 = j; }
    float e2 = __expf(v[i2] - v[i1]);
    float inv = 1.f / (1.f + e2);
    float g[4] = {0.f, 0.f, 0.f, 0.f};
    g[i1] = inv; g[i2] = e2 * inv;
    float* gp = gates + (size_t)node * NEXP;
    gp[0] = g[0]; gp[1] = g[1]; gp[2] = g[2]; gp[3] = g[3];
  }
}

// ---------------------------- edge scatter ----------------------------------
// One wave per edge; lanes cover 256 features in 8 strided steps. src row read
// and dst atomic adds are fully coalesced; agg (51MB) is L2-resident.

__global__ void scatter_kernel(const float* __restrict__ h, const float* __restrict__ nsrc,
                               const int* __restrict__ src, const int* __restrict__ dst,
                               float* __restrict__ agg) {
  int gid = blockIdx.x * blockDim.x + threadIdx.x;
  int e = gid >> 5;
  int lane = gid & 31;
  if (e >= NEDGE) return;
  int s = src[e], d = dst[e];
  float ns = nsrc[s];
  const float* hp = h + (size_t)s * DIM;
  float* ap = agg + (size_t)d * DIM;
#pragma unroll
  for (int j = 0; j < 8; j++) {
    int c = lane + j * 32;
    atomicAdd(&ap[c], hp[c] * ns);
  }
}

// ------------------------- WMMA expert GEMM core ----------------------------
// One wave: 16-node × 16-feature output tile for all 4 experts, K=256 in
// 16x16x4 fp32 WMMA steps. A from LDS (norm_dst pre-folded), B from L2, bias
// folded into the C accumulator init (all 8 C slots of a lane share one f).

__device__ __forceinline__ void stage_A(float* lA, const float* __restrict__ agg,
                                        const float* __restrict__ ndst, int node0, int tid) {
  int r = tid >> 3;
  int cc = (tid & 7) * 32;
  float nd = ndst[node0 + r];
  const float* srcp = agg + (size_t)(node0 + r) * DIM + cc;
  float* dstp = lA + r * LDA + cc;
#pragma unroll
  for (int j = 0; j < 8; j++) {
    float4 v = *(const float4*)(srcp + j * 4);
    v.x *= nd; v.y *= nd; v.z *= nd; v.w *= nd;
    *(float4*)(dstp + j * 4) = v;
  }
}

__device__ __forceinline__ void expert_gemm(const float* lA, const float* __restrict__ W,
                                            const float* __restrict__ bvec,
                                            int f0, int lane, v8f c[NEXP]) {
  const int half = lane >> 4;   // A: K-pair select; C: row-half select
  const int lm = lane & 15;     // A: row M; B/C: column N (= feature)
  const int f = f0 + lm;
#pragma unroll
  for (int e = 0; e < NEXP; e++) {
    float b0 = bvec[e * DIM + f];
    c[e] = (v8f){b0, b0, b0, b0, b0, b0, b0, b0};
  }
  const float* lArow = lA + lm * LDA + 2 * half;
#pragma unroll 4
  for (int k0 = 0; k0 < DIM; k0 += 4) {
    v2f a = *(const v2f*)(lArow + k0);                      // A[m][k0+2h], A[m][k0+2h+1]
    const float* wk = W + (size_t)(k0 + 2 * half) * DIM + f;
#pragma unroll
    for (int e = 0; e < NEXP; e++) {
      const float* wp = wk + (size_t)e * DIM * DIM;
      v2f b;
      b.x = wp[0];                                          // B[k0+2h][f]
      b.y = wp[DIM];                                        // B[k0+2h+1][f]
      c[e] = __builtin_amdgcn_wmma_f32_16x16x4_f32(
          false, a, false, b, (short)0, c[e], false, false);
    }
  }
}

// Pass A: GEMM -> per-(expert,feature) sum / sumsq (BN training stats).
__global__ void __launch_bounds__(128) moe_gemm_stats(
    const float* __restrict__ agg, const float* __restrict__ ndst,
    const float* __restrict__ W, const float* __restrict__ bvec,
    float* __restrict__ ssum, float* __restrict__ ssq) {
  __shared__ float lA[16 * LDA];
  int node0 = blockIdx.x * 16;
  int tid = threadIdx.x;
  stage_A(lA, agg, ndst, node0, tid);
  __syncthreads();
  int wave = tid >> 5, lane = tid & 31;
  int f0 = blockIdx.y * 64 + wave * 16;
  v8f c[NEXP];
  expert_gemm(lA, W, bvec, f0, lane, c);
  int f = f0 + (lane & 15);
#pragma unroll
  for (int e = 0; e < NEXP; e++) {
    float s = 0.f, q = 0.f;
#pragma unroll
    for (int v = 0; v < 8; v++) { float z = c[e][v]; s += z; q += z * z; }
    s += __shfl_xor(s, 16, 32);   // fold rows 8..15 into rows 0..7 lanes
    q += __shfl_xor(q, 16, 32);
    if (lane < 16) {
      atomicAdd(&ssum[e * DIM + f], s);
      atomicAdd(&ssq[e * DIM + f], q);
    }
  }
}

// scale = gamma * rsqrt(var+eps); shift = beta - mu*scale
__global__ void bn_finalize(const float* __restrict__ ssum, const float* __restrict__ ssq,
                            const float* __restrict__ gamma, const float* __restrict__ beta,
                            float* __restrict__ scl, float* __restrict__ shf) {
  int i = blockIdx.x * blockDim.x + threadIdx.x;
  if (i < NEXP * DIM) {
    float mu = ssum[i] * (1.f / (float)NNODES);
    float var = ssq[i] * (1.f / (float)NNODES) - mu * mu;
    float inv = rsqrtf(var + BN_EPS);
    float sc = gamma[i] * inv;
    scl[i] = sc;
    shf[i] = beta[i] - mu * sc;
  }
}

// Pass B: recompute z via WMMA, BN-normalize, gate-combine, h + relu(h).
__global__ void __launch_bounds__(128) moe_gemm_combine(
    const float* __restrict__ agg, const float* __restrict__ ndst,
    const float* __restrict__ W, const float* __restrict__ bvec,
    const float* __restrict__ scl, const float* __restrict__ shf,
    const float* __restrict__ gates, float* __restrict__ hout) {
  __shared__ float lA[16 * LDA];
  __shared__ float lG[16 * NEXP];
  int node0 = blockIdx.x * 16;
  int tid = threadIdx.x;
  stage_A(lA, agg, ndst, node0, tid);
  if (tid < 16 * NEXP) lG[tid] = gates[(size_t)node0 * NEXP + tid];
  __syncthreads();
  int wave = tid >> 5, lane = tid & 31;
  int f0 = blockIdx.y * 64 + wave * 16;
  v8f c[NEXP];
  expert_gemm(lA, W, bvec, f0, lane, c);
  int half = lane >> 4, lm = lane & 15;
  int f = f0 + lm;
  float sc[NEXP], sh[NEXP];
#pragma unroll
  for (int e = 0; e < NEXP; e++) { sc[e] = scl[e * DIM + f]; sh[e] = shf[e * DIM + f]; }
#pragma unroll
  for (int v = 0; v < 8; v++) {
    int m = v + 8 * half;                    // row within node tile
    float acc = 0.f;
#pragma unroll
    for (int e = 0; e < NEXP; e++)
      acc += lG[m * NEXP + e] * (c[e][v] * sc[e] + sh[e]);
    float outv = acc + fmaxf(acc, 0.f);      // h_moe + relu(h_moe)
    hout[(size_t)(node0 + m) * DIM + f] = outv;
  }
}

// ------------------------------ orchestration -------------------------------

extern "C" void kernel_launch(void* const* d_in, const int* in_sizes, int n_in,
                              void* d_out, int out_size, void* d_ws, size_t ws_size,
                              hipStream_t stream) {
  const int*   x   = (const int*)d_in[0];
  const int*   ei  = (const int*)d_in[1];
  const float* emb = (const float*)d_in[2];
  const float* wg  = (const float*)d_in[3];   // [L][D][E]
  const float* cW  = (const float*)d_in[4];   // [L][E][D][D]
  const float* cb  = (const float*)d_in[5];   // [L][E][D]
  const float* gma = (const float*)d_in[6];
  const float* bta = (const float*)d_in[7];
  float* out = (float*)d_out;
  float* ws  = (float*)d_ws;

  const int* esrc = ei;
  const int* edst = ei + NEDGE;

  size_t off = 0;
  float* outdeg = ws + off; off += NNODES;
  float* indeg  = ws + off; off += NNODES;
  float* nsrc   = ws + off; off += NNODES;
  float* ndst   = ws + off; off += NNODES;
  float* hA     = ws + off; off += (size_t)NNODES * DIM;
  float* hB     = ws + off; off += (size_t)NNODES * DIM;
  float* agg    = ws + off; off += (size_t)NNODES * DIM;
  float* gates  = ws + off; off += (size_t)NNODES * NEXP;
  float* ssum   = ws + off; off += NEXP * DIM;
  float* ssq    = ws + off; off += NEXP * DIM;   // contiguous with ssum for joint zero
  float* scl    = ws + off; off += NEXP * DIM;
  float* shf    = ws + off; off += NEXP * DIM;
  // total ~38.8M floats (~155MB) of d_ws

  zero_f<<<(2 * NNODES + 255) / 256, 256, 0, stream>>>(outdeg, 2 * NNODES);
  deg_kernel<<<(NEDGE + 255) / 256, 256, 0, stream>>>(esrc, edst, outdeg, indeg);
  norm_kernel<<<(NNODES + 255) / 256, 256, 0, stream>>>(outdeg, indeg, nsrc, ndst);
  embed_kernel<<<(NNODES * 64 + 255) / 256, 256, 0, stream>>>(x, emb, hA);

  float* hin = hA;
  for (int l = 0; l < NLAYER; l++) {
    float* hout = (l == NLAYER - 1) ? out : ((hin == hA) ? hB : hA);
    const float* Wl = cW + (size_t)l * NEXP * DIM * DIM;
    const float* bl = cb + (size_t)l * NEXP * DIM;

    gate_kernel<<<(NNODES * 32 + 255) / 256, 256, 0, stream>>>(
        hin, wg + (size_t)l * DIM * NEXP, gates);
    zero_f<<<(NNODES * DIM + 255) / 256, 256, 0, stream>>>(agg, NNODES * DIM);
    zero_f<<<(2 * NEXP * DIM + 255) / 256, 256, 0, stream>>>(ssum, 2 * NEXP * DIM);
    scatter_kernel<<<(NEDGE * 32 + 255) / 256, 256, 0, stream>>>(hin, nsrc, esrc, edst, agg);

    dim3 g(NNODES / 16, 4);   // 3125 node tiles x 4 feature groups (4 waves x 16f each)
    moe_gemm_stats<<<g, 128, 0, stream>>>(agg, ndst, Wl, bl, ssum, ssq);
    bn_finalize<<<(NEXP * DIM + 255) / 256, 256, 0, stream>>>(
        ssum, ssq, gma + (size_t)l * NEXP * DIM, bta + (size_t)l * NEXP * DIM, scl, shf);
    moe_gemm_combine<<<g, 128, 0, stream>>>(agg, ndst, Wl, bl, scl, shf, gates, hout);

    hin = hout;
  }
  (void)in_sizes; (void)n_in; (void)out_size; (void)ws_size;
}